// LocSE_11725260718183
// MI455X (gfx1250) — compile-verified
//
#include <hip/hip_runtime.h>

typedef __attribute__((ext_vector_type(2))) float v2f;
typedef __attribute__((ext_vector_type(4))) float f4;
typedef __attribute__((ext_vector_type(8))) float v8f;

#define NT 256
typedef unsigned long long u64;
typedef unsigned int u32;

// ---------------------------------------------------------------------------
// Register-resident top-16 of packed keys. key = (dist_bits << 32) | ~idx:
// larger dist -> larger key; equal dist -> smaller idx wins (top_k stability).
// Fully unrolled so kk[] stays in VGPRs (no scratch spills from dyn indexing).
// ---------------------------------------------------------------------------
__device__ __forceinline__ void topk_insert(u64 (&kk)[16], u64& mn, int& mnPos, u64 key) {
    if (key > mn) {
#pragma unroll
        for (int c = 0; c < 16; ++c)
            if (c == mnPos) kk[c] = key;
        mn = kk[0]; mnPos = 0;
#pragma unroll
        for (int c = 1; c < 16; ++c)
            if (kk[c] < mn) { mn = kk[c]; mnPos = c; }
    }
}

__device__ __forceinline__ u64 make_key(float d, u32 idx) {
    return ((u64)__float_as_uint(d) << 32) | (u32)(~idx);
}

// ---------------------------------------------------------------------------
// Kernel 1: streaming distance scan + per-block top-16.
// Each thread handles 2 consecutive rows per iteration: 48B = 3 x b128 loads,
// 16B-aligned (row pair starts at 48B boundary), non-temporal (pure stream).
// Block reduction: 16 passes of LDS atomicMax (ds_max_u64); keys are unique
// so exactly one owner clears its slot each pass.
// ---------------------------------------------------------------------------
__global__ void __launch_bounds__(NT) k_scan(const float* __restrict__ P,
                                             const int* __restrict__ ip,
                                             u64* __restrict__ cand, int n) {
    const int i = ip[0];
    const float pix = P[(size_t)i * 6 + 0];
    const float piy = P[(size_t)i * 6 + 1];
    const float piz = P[(size_t)i * 6 + 2];

    u64 kk[16];
#pragma unroll
    for (int c = 0; c < 16; ++c) kk[c] = 0ull;
    u64 mn = 0ull; int mnPos = 0;

    const int tid    = blockIdx.x * NT + threadIdx.x;
    const int stride = gridDim.x * NT;

    for (int base = tid * 2; base < n; base += stride * 2) {
        if (base + 1 < n) {
            const f4* rp = (const f4*)(P + (size_t)base * 6);
            f4 q0 = __builtin_nontemporal_load(rp + 0);
            f4 q1 = __builtin_nontemporal_load(rp + 1);
            f4 q2 = __builtin_nontemporal_load(rp + 2);
            // row base:   xyz = q0.x q0.y q0.z
            float dx0 = q0.x - pix, dy0 = q0.y - piy, dz0 = q0.z - piz;
            float d0  = sqrtf(dx0 * dx0 + dy0 * dy0 + dz0 * dz0);
            topk_insert(kk, mn, mnPos, make_key(d0, (u32)base));
            // row base+1: xyz = q1.z q1.w q2.x
            float dx1 = q1.z - pix, dy1 = q1.w - piy, dz1 = q2.x - piz;
            float d1  = sqrtf(dx1 * dx1 + dy1 * dy1 + dz1 * dz1);
            topk_insert(kk, mn, mnPos, make_key(d1, (u32)(base + 1)));
        } else {
            const float* rp = P + (size_t)base * 6;
            float dx = rp[0] - pix, dy = rp[1] - piy, dz = rp[2] - piz;
            float d  = sqrtf(dx * dx + dy * dy + dz * dz);
            topk_insert(kk, mn, mnPos, make_key(d, (u32)base));
        }
    }

    // Block-level top-16 via LDS u64 atomic-max passes.
    __shared__ u64 gmax;
    for (int p = 0; p < 16; ++p) {
        if (threadIdx.x == 0) gmax = 0ull;
        __syncthreads();
        u64 lm = kk[0];
#pragma unroll
        for (int c = 1; c < 16; ++c) lm = (kk[c] > lm) ? kk[c] : lm;
        if (lm) atomicMax(&gmax, lm);
        __syncthreads();
        u64 g = gmax;
        if (g && lm == g) {   // unique owner: remove winner from its set
#pragma unroll
            for (int c = 0; c < 16; ++c)
                if (kk[c] == g) kk[c] = 0ull;
        }
        if (threadIdx.x == 0) cand[(size_t)blockIdx.x * 16 + p] = g;
        __syncthreads();
    }
}

// ---------------------------------------------------------------------------
// Kernel 2: one block. Reduce nb*16 candidates to global top-16 (descending),
// gather rows, build feat (16x10 padded to 16x12) in LDS, then GEMM with
// three chained V_WMMA_F32_16X16X4_F32 (full fp32 precision, K padded 12).
// ---------------------------------------------------------------------------
__global__ void __launch_bounds__(NT) k_final(const float* __restrict__ P,
                                              const float* __restrict__ W,
                                              const float* __restrict__ bias,
                                              const int* __restrict__ ip,
                                              const u64* __restrict__ cand, int ncand,
                                              float* __restrict__ out) {
    __shared__ u64   gmax;
    __shared__ u64   sTop[16];
    __shared__ float sFeat[16][12];
    __shared__ float sD[16][16];

    // --- per-thread top-16 over strided candidate list ---
    u64 kk[16];
#pragma unroll
    for (int c = 0; c < 16; ++c) kk[c] = 0ull;
    u64 mn = 0ull; int mnPos = 0;
    for (int c = threadIdx.x; c < ncand; c += NT)
        topk_insert(kk, mn, mnPos, cand[c]);

    // --- 16 selection passes -> sTop in descending order ---
    for (int p = 0; p < 16; ++p) {
        if (threadIdx.x == 0) gmax = 0ull;
        __syncthreads();
        u64 lm = kk[0];
#pragma unroll
        for (int c = 1; c < 16; ++c) lm = (kk[c] > lm) ? kk[c] : lm;
        if (lm) atomicMax(&gmax, lm);
        __syncthreads();
        u64 g = gmax;
        if (g && lm == g) {
#pragma unroll
            for (int c = 0; c < 16; ++c)
                if (kk[c] == g) kk[c] = 0ull;
        }
        if (threadIdx.x == 0) sTop[p] = g;
        __syncthreads();
    }

    const int i = ip[0];
    const float pix = P[(size_t)i * 6 + 0];
    const float piy = P[(size_t)i * 6 + 1];
    const float piz = P[(size_t)i * 6 + 2];

    // --- gather rows, build feat, emit nloc part of output ---
    if (threadIdx.x < 16) {
        const int r = threadIdx.x;
        u32 idx = ~(u32)(sTop[r] & 0xffffffffu);
        const float* rp = P + (size_t)idx * 6;
        float nx = rp[0], ny = rp[1], nz = rp[2];
        float dx = pix - nx, dy = piy - ny, dz = piz - nz;
        float dist = sqrtf(dx * dx + dy * dy + dz * dz);
        sFeat[r][0] = pix;  sFeat[r][1] = piy;  sFeat[r][2] = piz;
        sFeat[r][3] = nx;   sFeat[r][4] = ny;   sFeat[r][5] = nz;
        sFeat[r][6] = dx;   sFeat[r][7] = dy;   sFeat[r][8] = dz;
        sFeat[r][9] = dist; sFeat[r][10] = 0.f; sFeat[r][11] = 0.f;
        out[r * 6 + 0] = nx; out[r * 6 + 1] = ny; out[r * 6 + 2] = nz;
    }
    __syncthreads();

    // --- wave 0 only (EXEC all-1s within the wave): R = feat @ W^T ---
    if (threadIdx.x < 32) {
        const int L = threadIdx.x;
        const int half = L >> 4;      // 0: K pair {0,1}; 1: K pair {2,3}
        const int r = L & 15;         // A row / B column
        v8f acc = {};
#pragma unroll
        for (int c2 = 0; c2 < 3; ++c2) {
            const int ka = c2 * 4 + 2 * half;
            // A (16x4 f32): lane r holds feat[r][ka], feat[r][ka+1]
            v2f a;
            a.x = sFeat[r][ka];
            a.y = sFeat[r][ka + 1];
            // B (4x16 f32): B[k][n] = W^T[k][n] = W[n][k], zero-padded
            float b0 = 0.f, b1 = 0.f;
            if (r < 3) {
                if (ka + 0 < 10) b0 = W[r * 10 + ka + 0];
                if (ka + 1 < 10) b1 = W[r * 10 + ka + 1];
            }
            v2f bmat; bmat.x = b0; bmat.y = b1;
            acc = __builtin_amdgcn_wmma_f32_16x16x4_f32(
                /*neg_a=*/false, a, /*neg_b=*/false, bmat,
                /*c_mod=*/(short)0, acc, /*reuse_a=*/false, /*reuse_b=*/false);
        }
        // D layout: VGPR v, low half = row v, high half = row v+8, col = r
#pragma unroll
        for (int v = 0; v < 8; ++v) sD[v + 8 * half][r] = acc[v];
    }
    __syncthreads();

    if (threadIdx.x < 16) {
        const int r = threadIdx.x;
        out[r * 6 + 3] = sD[r][0] + bias[0];
        out[r * 6 + 4] = sD[r][1] + bias[1];
        out[r * 6 + 5] = sD[r][2] + bias[2];
    }
}

extern "C" void kernel_launch(void* const* d_in, const int* in_sizes, int n_in,
                              void* d_out, int out_size, void* d_ws, size_t ws_size,
                              hipStream_t stream) {
    const float* P    = (const float*)d_in[0];
    const float* W    = (const float*)d_in[1];
    const float* bias = (const float*)d_in[2];
    const int*   ip   = (const int*)d_in[3];
    float* out        = (float*)d_out;

    const int n = in_sizes[0] / 6;

    // workspace: nb * 16 packed u64 candidate keys
    int nb = (int)(ws_size / (16 * sizeof(u64)));
    if (nb > 512) nb = 512;
    if (nb < 1) nb = 1;
    u64* cand = (u64*)d_ws;

    k_scan<<<nb, NT, 0, stream>>>(P, ip, cand, n);
    k_final<<<1, NT, 0, stream>>>(P, W, bias, ip, cand, nb * 16, out);
}